// MultiheadAttention_19791209300557
// MI455X (gfx1250) — compile-verified
//
#include <hip/hip_runtime.h>

typedef float v2f __attribute__((ext_vector_type(2)));
typedef float v8f __attribute__((ext_vector_type(8)));
typedef unsigned int v4u __attribute__((ext_vector_type(4)));
typedef int v4i __attribute__((ext_vector_type(4)));
typedef int v8i __attribute__((ext_vector_type(8)));

#define D_MODEL   1024
#define NUM_HEADS 16
#define D_HEAD    64
#define SEQ       2048
#define BATCH     2

static __device__ __forceinline__ v8f v8f_zero() {
  v8f z = {0.f, 0.f, 0.f, 0.f, 0.f, 0.f, 0.f, 0.f};
  return z;
}

// D = A(16x4 f32) * B(4x16 f32) + C(16x16 f32), full fp32 matrix pipe.
static __device__ __forceinline__ v8f wmma4(v2f a, v2f b, v8f c) {
  return __builtin_amdgcn_wmma_f32_16x16x4_f32(
      /*neg_a=*/false, a, /*neg_b=*/false, b,
      /*c_mod=*/(short)0, c, /*reuse_a=*/false, /*reuse_b=*/false);
}

// ---------------------------------------------------------------------------
// TDM: DMA one tileW x tileH fp32 tile (memory row stride = strideElems) into
// LDS, with hardware row padding (pad_amount DWORDs appended per pad_interval
// DWORDs stored) so LDS rows land at a bank-conflict-free stride.
//   padIntervalCode: 2^(c+1) DWORDs between pads (4 -> 32 DW, 5 -> 64 DW)
//   padAmountCode:   c+1 DWORDs of pad            (3 -> 4 DW)
// D# packing per CDNA5 ISA ch.8. All inputs are workgroup-uniform -> SGPRs.
// ---------------------------------------------------------------------------
static __device__ __forceinline__ void tdm_load_tile(
    unsigned ldsAddr, const float* gaddr, unsigned tileW, unsigned tileH,
    unsigned strideElems, unsigned padIntervalCode, unsigned padAmountCode) {
  const unsigned long long ga = (unsigned long long)(size_t)gaddr;
  v4u g0;
  g0.x = 1u;                                            // count=1, user mode
  g0.y = ldsAddr;                                       // lds_addr (bytes)
  g0.z = (unsigned)(ga & 0xffffffffu);                  // global_addr[31:0]
  g0.w = (unsigned)((ga >> 32) & 0x01ffffffu) | (2u << 30);  // [56:32] | type=2
  v8i g1;
  // data_size=4B | pad_enable | pad_interval | pad_amount; wg_mask=0
  g1[0] = (int)((2u << 16) | (1u << 20) | (padIntervalCode << 22) |
                (padAmountCode << 25));
  g1[1] = (int)(tileW << 16);   // tensor_dim0 (atomic_barrier_addr = 0)
  g1[2] = (int)(tileH << 16);   // tensor_dim0 hi=0 | tensor_dim1 lo
  g1[3] = (int)(tileW << 16);   // tensor_dim1 hi=0 | tile_dim0
  g1[4] = (int)tileH;           // tile_dim1 | tile_dim2=0
  g1[5] = (int)strideElems;     // tensor_dim0_stride lo32 (elems)
  g1[6] = 0;                    // stride0 hi / stride1 lo
  g1[7] = 0;                    // stride1 hi
  v4i z4 = {0, 0, 0, 0};
#if __clang_major__ >= 23
  v8i z8 = {0, 0, 0, 0, 0, 0, 0, 0};
  __builtin_amdgcn_tensor_load_to_lds(g0, g1, z4, z4, z8, 0);
#else
  __builtin_amdgcn_tensor_load_to_lds(g0, g1, z4, z4, 0);
#endif
}

// ---------------------------------------------------------------------------
// Kernel 1: fused QKV projection.  Y = X @ W^T + bias, written as [B,H,L,Dh].
// blockIdx.z in {0,1,2} selects the Q/K/V weight set.
// Tile: 128 rows x 128 cols per 256-thread block; the 128x32 X/W K-panels are
// double-buffered in LDS and staged by the TDM (DMA of step kk+1 overlaps the
// 64 WMMAs of step kk; one barrier per step).
// ---------------------------------------------------------------------------
__global__ __launch_bounds__(256) void qkv_proj_kernel(
    const float* __restrict__ xq, const float* __restrict__ xk, const float* __restrict__ xv,
    const float* __restrict__ wq, const float* __restrict__ wk, const float* __restrict__ wv,
    const float* __restrict__ bq, const float* __restrict__ bk, const float* __restrict__ bv,
    float* __restrict__ oq, float* __restrict__ ok, float* __restrict__ ov) {
  const int which = blockIdx.z;
  const float* __restrict__ x    = (which == 0) ? xq : (which == 1) ? xk : xv;
  const float* __restrict__ w    = (which == 0) ? wq : (which == 1) ? wk : wv;
  const float* __restrict__ bias = (which == 0) ? bq : (which == 1) ? bk : bv;
  float* __restrict__ o          = (which == 0) ? oq : (which == 1) ? ok : ov;

  __shared__ float Xs[2][128][36];  // TDM pads 32->36 DW rows (conflict-free)
  __shared__ float Ws[2][128][36];

  const int tid  = threadIdx.x;
  const int lane = tid & 31;
  const int wave = tid >> 5;     // 8 waves, each owns 16 output rows
  const int nl   = lane & 15;    // N (or M for A-frag) index within half-wave
  const int lh   = lane >> 4;    // half-wave select -> K pair / row+8

  const int rowBase = blockIdx.x * 128;
  const int colBase = blockIdx.y * 128;
  const float* xBase = x + (size_t)rowBase * D_MODEL;
  const float* wBase = w + (size_t)colBase * D_MODEL;

  v8f acc[8];
#pragma unroll
  for (int t = 0; t < 8; ++t) acc[t] = v8f_zero();

  // Prologue: DMA K-panel 0
  if (wave == 0) {
    tdm_load_tile((unsigned)(size_t)&Xs[0][0][0], xBase, 32, 128, D_MODEL, 4, 3);
    tdm_load_tile((unsigned)(size_t)&Ws[0][0][0], wBase, 32, 128, D_MODEL, 4, 3);
    __builtin_amdgcn_s_wait_tensorcnt(0);
  }
  __syncthreads();

  const int NKT = D_MODEL / 32;
  for (int it = 0; it < NKT; ++it) {
    const int cur = it & 1, nxt = cur ^ 1;
    if (wave == 0 && it + 1 < NKT) {
      tdm_load_tile((unsigned)(size_t)&Xs[nxt][0][0], xBase + (it + 1) * 32,
                    32, 128, D_MODEL, 4, 3);
      tdm_load_tile((unsigned)(size_t)&Ws[nxt][0][0], wBase + (it + 1) * 32,
                    32, 128, D_MODEL, 4, 3);
    }

#pragma unroll
    for (int s = 0; s < 8; ++s) {
      const int ks = s * 4;
      // A frag 16x4: lane m=nl, element e holds K = ks + e + 2*lh
      v2f a;
      a.x = Xs[cur][wave * 16 + nl][ks + 2 * lh];
      a.y = Xs[cur][wave * 16 + nl][ks + 2 * lh + 1];
#pragma unroll
      for (int t = 0; t < 8; ++t) {
        // B frag 4x16: B[k][n] = W[colBase + t*16 + n][k]
        v2f bf;
        bf.x = Ws[cur][t * 16 + nl][ks + 2 * lh];
        bf.y = Ws[cur][t * 16 + nl][ks + 2 * lh + 1];
        acc[t] = wmma4(a, bf, acc[t]);
      }
    }

    if (wave == 0) __builtin_amdgcn_s_wait_tensorcnt(0);
    __syncthreads();
  }

  // Epilogue: + bias, scatter into [B, H, L, Dh]
#pragma unroll
  for (int t = 0; t < 8; ++t) {
    const int n  = colBase + t * 16 + nl;
    const float bb = bias[n];
    const int h = n >> 6, dh = n & 63;
#pragma unroll
    for (int vr = 0; vr < 8; ++vr) {
      const int m = rowBase + wave * 16 + vr + 8 * lh;  // global row = b*L + l
      const int b = m >> 11, l = m & (SEQ - 1);
      o[(((size_t)b * NUM_HEADS + h) * SEQ + l) * D_HEAD + dh] = acc[t][vr] + bb;
    }
  }
}

// ---------------------------------------------------------------------------
// Kernel 2: flash attention, fp32 WMMA, online softmax.
// One block per (b,h, 128-row Q block); 8 waves x 16 rows each.
// Q and the double-buffered K/V tiles are all staged by the TDM; wave 0
// issues DMA for tile kt+1 while all waves compute tile kt.
// ---------------------------------------------------------------------------
__global__ __launch_bounds__(256) void flash_attn_kernel(
    const float* __restrict__ q, const float* __restrict__ k,
    const float* __restrict__ v, float* __restrict__ out) {
  __shared__ float Qs[128][68];    // TDM pads 64->68 DW rows
  __shared__ float Ks[2][64][68];
  __shared__ float Vs[2][64][68];
  __shared__ float Ps[8][16][68];  // per-wave P staging (C-layout -> A-layout)

  const int tid  = threadIdx.x;
  const int lane = tid & 31;
  const int wave = tid >> 5;
  const int nl   = lane & 15;
  const int lh   = lane >> 4;

  const int bh      = blockIdx.y;            // b*NUM_HEADS + h
  const int qBase   = blockIdx.x * 128;
  const size_t headOff = (size_t)bh * SEQ * D_HEAD;

  // DMA the whole 128x64 Q tile + K/V tile 0; one wait covers all three.
  if (wave == 0) {
    tdm_load_tile((unsigned)(size_t)&Qs[0][0],
                  q + headOff + (size_t)qBase * D_HEAD, 64, 128, 64, 5, 3);
    tdm_load_tile((unsigned)(size_t)&Ks[0][0][0], k + headOff, 64, 64, 64, 5, 3);
    tdm_load_tile((unsigned)(size_t)&Vs[0][0][0], v + headOff, 64, 64, 64, 5, 3);
    __builtin_amdgcn_s_wait_tensorcnt(0);
  }
  __syncthreads();

  // This wave's 16xDh Q slice as 16 A-fragments, scaled by 1/sqrt(Dh) = 1/8
  v2f qf[16];
#pragma unroll
  for (int j = 0; j < 16; ++j) {
    qf[j].x = 0.125f * Qs[wave * 16 + nl][j * 4 + 2 * lh];
    qf[j].y = 0.125f * Qs[wave * 16 + nl][j * 4 + 2 * lh + 1];
  }

  v8f acc[4];
#pragma unroll
  for (int t = 0; t < 4; ++t) acc[t] = v8f_zero();
  float m_i[8], l_i[8];
#pragma unroll
  for (int r = 0; r < 8; ++r) { m_i[r] = -1e30f; l_i[r] = 0.f; }

  const int NT = SEQ / 64;
  for (int kt = 0; kt < NT; ++kt) {
    const int cur = kt & 1, nxt = cur ^ 1;

    // Prefetch next K/V tile into the other buffer (overlaps with compute)
    if (wave == 0 && kt + 1 < NT) {
      const size_t off = headOff + (size_t)(kt + 1) * 64 * D_HEAD;
      tdm_load_tile((unsigned)(size_t)&Ks[nxt][0][0], k + off, 64, 64, 64, 5, 3);
      tdm_load_tile((unsigned)(size_t)&Vs[nxt][0][0], v + off, 64, 64, 64, 5, 3);
    }

    // S = (Q/8) @ K^T : 16x64 tile per wave, C-layout
    v8f s[4];
#pragma unroll
    for (int t = 0; t < 4; ++t) s[t] = v8f_zero();
#pragma unroll
    for (int j = 0; j < 16; ++j) {
      v2f a = qf[j];
#pragma unroll
      for (int t = 0; t < 4; ++t) {
        v2f bf;  // B[k][n] = K[kt*64 + t*16 + n][k]
        bf.x = Ks[cur][t * 16 + nl][j * 4 + 2 * lh];
        bf.y = Ks[cur][t * 16 + nl][j * 4 + 2 * lh + 1];
        s[t] = wmma4(a, bf, s[t]);
      }
    }

    // Online softmax. C-layout: VGPR vr <-> row vr + 8*lh; columns across the
    // 16 lanes of each half-wave. xor-shuffles of 1,2,4,8 stay in-half.
#pragma unroll
    for (int vr = 0; vr < 8; ++vr) {
      float rowm = fmaxf(fmaxf(s[0][vr], s[1][vr]), fmaxf(s[2][vr], s[3][vr]));
#pragma unroll
      for (int off = 1; off < 16; off <<= 1)
        rowm = fmaxf(rowm, __shfl_xor(rowm, off, 32));
      const float m_new = fmaxf(m_i[vr], rowm);
      const float corr  = __expf(m_i[vr] - m_new);
      float rowsum = 0.f;
#pragma unroll
      for (int t = 0; t < 4; ++t) {
        float p = __expf(s[t][vr] - m_new);
        s[t][vr] = p;
        rowsum += p;
      }
#pragma unroll
      for (int off = 1; off < 16; off <<= 1)
        rowsum += __shfl_xor(rowsum, off, 32);
      l_i[vr] = l_i[vr] * corr + rowsum;
      m_i[vr] = m_new;
#pragma unroll
      for (int t = 0; t < 4; ++t) acc[t][vr] *= corr;
    }

    // C-layout -> A-layout via per-wave LDS pad (wave-private, DS in-order)
#pragma unroll
    for (int t = 0; t < 4; ++t)
#pragma unroll
      for (int vr = 0; vr < 8; ++vr)
        Ps[wave][vr + 8 * lh][t * 16 + nl] = s[t][vr];

    // O += P @ V
#pragma unroll
    for (int j = 0; j < 16; ++j) {
      v2f a;
      a.x = Ps[wave][nl][j * 4 + 2 * lh];
      a.y = Ps[wave][nl][j * 4 + 2 * lh + 1];
#pragma unroll
      for (int t = 0; t < 4; ++t) {
        v2f bf;  // B[k][n] = V[kt*64 + k][t*16 + n]
        bf.x = Vs[cur][j * 4 + 2 * lh][t * 16 + nl];
        bf.y = Vs[cur][j * 4 + 2 * lh + 1][t * 16 + nl];
        acc[t] = wmma4(a, bf, acc[t]);
      }
    }

    // Next tile's DMA must be complete and everyone done with `cur` buffer
    if (wave == 0) __builtin_amdgcn_s_wait_tensorcnt(0);
    __syncthreads();
  }

  // Normalize and write out [B, L, E] (heads concatenated)
  const int b = bh >> 4, h = bh & 15;
#pragma unroll
  for (int vr = 0; vr < 8; ++vr) {
    const float inv = 1.f / l_i[vr];
    const int l = qBase + wave * 16 + vr + 8 * lh;
#pragma unroll
    for (int t = 0; t < 4; ++t) {
      out[((size_t)b * SEQ + l) * D_MODEL + h * D_HEAD + t * 16 + nl] =
          acc[t][vr] * inv;
    }
  }
}

// ---------------------------------------------------------------------------
extern "C" void kernel_launch(void* const* d_in, const int* in_sizes, int n_in,
                              void* d_out, int out_size, void* d_ws, size_t ws_size,
                              hipStream_t stream) {
  (void)in_sizes; (void)n_in; (void)out_size; (void)ws_size;
  const float* query = (const float*)d_in[0];
  const float* key   = (const float*)d_in[1];
  const float* value = (const float*)d_in[2];
  const float* wq    = (const float*)d_in[3];
  const float* bq    = (const float*)d_in[4];
  const float* wk    = (const float*)d_in[5];
  const float* bk    = (const float*)d_in[6];
  const float* wv    = (const float*)d_in[7];
  const float* bv    = (const float*)d_in[8];
  float* out = (float*)d_out;

  const size_t tensorElems = (size_t)BATCH * SEQ * D_MODEL;  // 4M floats each
  float* qp = (float*)d_ws;
  float* kp = qp + tensorElems;
  float* vp = kp + tensorElems;

  dim3 pgrid(BATCH * SEQ / 128, D_MODEL / 128, 3);
  qkv_proj_kernel<<<pgrid, 256, 0, stream>>>(query, key, value,
                                             wq, wk, wv, bq, bk, bv,
                                             qp, kp, vp);

  dim3 agrid(SEQ / 128, BATCH * NUM_HEADS);
  flash_attn_kernel<<<agrid, 256, 0, stream>>>(qp, kp, vp, out);
}